// PowerSVDLayer1_5884105195540
// MI455X (gfx1250) — compile-verified
//
#include <hip/hip_runtime.h>

// PowerSVD layer collapsed to: S = X X^T, Q0 = X P0  (pass 1, WMMA SYRK)
//   -> tiny per-batch 32x32/32x3 algebra building projector M = Q Q^T
//   -> Out = M X  (pass 2, WMMA GEMM). fp32 WMMA 16x16x4 throughout.
//
// Memory-bound problem: X = 102.8 MB (L2-resident on 192MB L2), out = 102.8MB,
// P0 = 9.6MB -> ~220MB of traffic @ 23.3 TB/s ~ 10us floor; 3.3 GFLOP fp32 is
// negligible, so fp32 WMMA 16x16x4 is the right matrix op.

typedef float v8f __attribute__((ext_vector_type(8)));
typedef float v2f __attribute__((ext_vector_type(2)));

#define BATCH 64
#define CCH   32
#define NPIX  12544   // 112*112
#define RANK  3
#define CHUNKS 16
#define CHUNK (NPIX / CHUNKS)   // 784, divisible by 16

__device__ __forceinline__ v8f wmma_k4(v2f a, v2f b, v8f c) {
  // D = A(16x4) * B(4x16) + C, fp32
  return __builtin_amdgcn_wmma_f32_16x16x4_f32(false, a, false, b,
                                               (short)0, c, false, false);
}

__global__ void k_zero(float* __restrict__ p, int n) {
  int i = blockIdx.x * blockDim.x + threadIdx.x;
  if (i < n) p[i] = 0.0f;
}

// ---------------- Pass 1: S[b] = X X^T (32x32), Q0[b] = X P0 (32x3) --------
__global__ void __launch_bounds__(128)
k_pass1(const float* __restrict__ x, const float* __restrict__ p0,
        float* __restrict__ S, float* __restrict__ Q0) {
  const int b    = blockIdx.x;
  const int lane = threadIdx.x & 31;
  const int w    = threadIdx.x >> 5;   // 4 waves: one 16x16 S tile each
  const int Mt   = w >> 1;
  const int Nt   = w & 1;
  const int col  = lane & 15;
  const int kofs = (lane >> 4) << 1;   // lanes 16-31 hold K = 2,3

  const float* xb = x  + (size_t)b * CCH * NPIX;
  const float* pb = p0 + (size_t)b * NPIX * RANK;

  // A fragment: rows Mt*16..  B fragment: rows Nt*16.. (same layout, SYRK)
  const float* pa  = xb + (size_t)(Mt * 16 + col) * NPIX + kofs;
  const float* pbm = xb + (size_t)(Nt * 16 + col) * NPIX + kofs;

  v8f accS = {};
  v8f accQ = {};

  const int n0 = blockIdx.y * CHUNK;
  const int n1 = n0 + CHUNK;

  if (Nt == 0) {
    // Waves 0,2: accumulate an S tile AND a Q0 tile. Branch hoisted out of
    // the loop so the inner loop has no EXEC save/restore; unroll 4 so the
    // compiler can clause the loads and overlap them with WMMA issue.
    // Lanes col>=3 contribute zero columns to the Q0 tile.
    const float* pp = (col < RANK) ? (pb + kofs * RANK + col) : nullptr;
#pragma unroll 4
    for (int n = n0; n < n1; n += 4) {
      v2f a  = *(const v2f*)(pa + n);
      v2f bm = *(const v2f*)(pbm + n);
      v2f bp;
      if (pp) {   // lane-divergent loads only; WMMAs below are uniform
        bp.x = pp[(size_t)n * RANK];
        bp.y = pp[(size_t)n * RANK + RANK];
      } else {
        bp.x = 0.0f; bp.y = 0.0f;
      }
      accS = wmma_k4(a, bm, accS);
      accQ = wmma_k4(a, bp, accQ);
    }
  } else {
    // Waves 1,3: S tile only.
#pragma unroll 4
    for (int n = n0; n < n1; n += 4) {
      v2f a  = *(const v2f*)(pa + n);
      v2f bm = *(const v2f*)(pbm + n);
      accS = wmma_k4(a, bm, accS);
    }
  }

  // C/D layout: VGPR i, lanes 0-15 -> M=i, lanes 16-31 -> M=8+i; N=lane%16
  float* Sb = S  + (size_t)b * (CCH * CCH);
  float* Qb = Q0 + (size_t)b * (CCH * RANK);
  const int mbase = Mt * 16 + ((lane >> 4) << 3);
#pragma unroll
  for (int i = 0; i < 8; ++i)
    atomicAdd(&Sb[(mbase + i) * CCH + Nt * 16 + col], accS[i]);
  if (Nt == 0 && col < RANK) {
#pragma unroll
    for (int i = 0; i < 8; ++i)
      atomicAdd(&Qb[(mbase + i) * RANK + col], accQ[i]);
  }
}

// ---------------- Middle: build projector M = Q Q^T per batch --------------
__device__ __forceinline__ void cholinv3(const float G[3][3], float Li[3][3]) {
  float L00 = sqrtf(fmaxf(G[0][0], 1e-30f));
  float L10 = G[1][0] / L00;
  float L20 = G[2][0] / L00;
  float L11 = sqrtf(fmaxf(G[1][1] - L10 * L10, 1e-30f));
  float L21 = (G[2][1] - L20 * L10) / L11;
  float L22 = sqrtf(fmaxf(G[2][2] - L20 * L20 - L21 * L21, 1e-30f));
  float i00 = 1.0f / L00, i11 = 1.0f / L11, i22 = 1.0f / L22;
  Li[0][0] = i00;              Li[0][1] = 0.0f; Li[0][2] = 0.0f;
  Li[1][0] = -L10 * i00 * i11; Li[1][1] = i11;  Li[1][2] = 0.0f;
  Li[2][0] = (L10 * L21 * i11 - L20) * i00 * i22;
  Li[2][1] = -L21 * i11 * i22; Li[2][2] = i22;
}

__global__ void k_middle(const float* __restrict__ S, const float* __restrict__ Q0,
                         float* __restrict__ Mo) {
  int b = blockIdx.x * blockDim.x + threadIdx.x;
  if (b >= BATCH) return;
  const float* Sb = S  + (size_t)b * (CCH * CCH);
  const float* Qb = Q0 + (size_t)b * (CCH * RANK);

  // U = S * Q0   (32x3)
  float U[CCH][RANK];
  for (int i = 0; i < CCH; ++i) {
    float u0 = 0.f, u1 = 0.f, u2 = 0.f;
    for (int j = 0; j < CCH; ++j) {
      float s = Sb[i * CCH + j];
      u0 += s * Qb[j * RANK + 0];
      u1 += s * Qb[j * RANK + 1];
      u2 += s * Qb[j * RANK + 2];
    }
    U[i][0] = u0; U[i][1] = u1; U[i][2] = u2;
  }
  // G1 = Q0^T U = (X^T Q0)^T (X^T Q0); cholQR of P1 folded into A1 = L1^-T
  float G[3][3] = {};
  for (int i = 0; i < CCH; ++i)
    for (int r = 0; r < 3; ++r)
      for (int c = 0; c < 3; ++c)
        G[r][c] += Qb[i * RANK + r] * U[i][c];
  float Li[3][3];
  cholinv3(G, Li);
  // Q1 = U * Li^T  (== X * qr(P1).Q up to column signs, which cancel)
  float Q1[CCH][RANK];
  for (int i = 0; i < CCH; ++i)
    for (int r = 0; r < 3; ++r) {
      float v = 0.f;
      for (int k = 0; k <= r; ++k) v += U[i][k] * Li[r][k];
      Q1[i][r] = v;
    }
  // cholQR of Q1 -> orthonormal Q (32x3)
  float G2[3][3] = {};
  for (int i = 0; i < CCH; ++i)
    for (int r = 0; r < 3; ++r)
      for (int c = 0; c < 3; ++c)
        G2[r][c] += Q1[i][r] * Q1[i][c];
  float Li2[3][3];
  cholinv3(G2, Li2);
  float Q[CCH][RANK];
  for (int i = 0; i < CCH; ++i)
    for (int r = 0; r < 3; ++r) {
      float v = 0.f;
      for (int k = 0; k <= r; ++k) v += Q1[i][k] * Li2[r][k];
      Q[i][r] = v;
    }
  // M = Q Q^T (projector); Out = M X == Q (X^T Q)^T == reference output
  float* Mb = Mo + (size_t)b * (CCH * CCH);
  for (int i = 0; i < CCH; ++i)
    for (int j = 0; j < CCH; ++j) {
      float v = 0.f;
      for (int r = 0; r < 3; ++r) v += Q[i][r] * Q[j][r];
      Mb[i * CCH + j] = v;
    }
}

// ---------------- Pass 2: Out = M X (32x32 x 32xN GEMM) --------------------
__global__ void __launch_bounds__(128)
k_pass2(const float* __restrict__ x, const float* __restrict__ Mw,
        float* __restrict__ out) {
  const int b    = blockIdx.x;
  const int lane = threadIdx.x & 31;
  const int w    = threadIdx.x >> 5;
  const int col  = lane & 15;
  const int kofs = (lane >> 4) << 1;

  const float* xb = x  + (size_t)b * CCH * NPIX;
  const float* Mb = Mw + (size_t)b * CCH * CCH;
  float*       ob = out + (size_t)b * CCH * NPIX;

  // Preload all A fragments of M (2 M-tiles x 8 K-steps); M is L2-hot.
  v2f afrag[2][8];
#pragma unroll
  for (int Mt = 0; Mt < 2; ++Mt)
#pragma unroll
    for (int kb = 0; kb < 8; ++kb)
      afrag[Mt][kb] =
          *(const v2f*)(Mb + (size_t)(Mt * 16 + col) * CCH + kb * 4 + kofs);

  // Each WG covers 16 n-tiles; each wave 4 consecutive tiles.
  const int tile0 = blockIdx.y * 16 + w * 4;
#pragma unroll
  for (int t = 0; t < 4; ++t) {
    const int n = (tile0 + t) * 16;
    v2f bfrag[8];
#pragma unroll
    for (int kb = 0; kb < 8; ++kb) {   // B[k][j] = X[kb*4+k, n+j]
      const float* px = xb + (size_t)(kb * 4 + kofs) * NPIX + n + col;
      bfrag[kb].x = px[0];
      bfrag[kb].y = px[NPIX];
    }
    v8f acc0 = {}, acc1 = {};
#pragma unroll
    for (int kb = 0; kb < 8; ++kb) {
      acc0 = wmma_k4(afrag[0][kb], bfrag[kb], acc0);
      acc1 = wmma_k4(afrag[1][kb], bfrag[kb], acc1);
    }
    const int mbase = (lane >> 4) << 3;
#pragma unroll
    for (int i = 0; i < 8; ++i) {
      ob[(size_t)(mbase + i) * NPIX + n + col]      = acc0[i];
      ob[(size_t)(16 + mbase + i) * NPIX + n + col] = acc1[i];
    }
  }
}

extern "C" void kernel_launch(void* const* d_in, const int* in_sizes, int n_in,
                              void* d_out, int out_size, void* d_ws, size_t ws_size,
                              hipStream_t stream) {
  (void)in_sizes; (void)n_in; (void)out_size; (void)ws_size;
  const float* x  = (const float*)d_in[0];   // [B,C,H,W] fp32
  const float* p0 = (const float*)d_in[1];   // [B,N,3]   fp32
  float* out = (float*)d_out;

  float* S  = (float*)d_ws;                  // [B,32,32]
  float* Q0 = S  + BATCH * CCH * CCH;        // [B,32,3]
  float* M  = Q0 + BATCH * CCH * RANK;       // [B,32,32]

  const int nzero = BATCH * CCH * CCH + BATCH * CCH * RANK; // S + Q0
  k_zero<<<(nzero + 255) / 256, 256, 0, stream>>>(S, nzero);
  k_pass1<<<dim3(BATCH, CHUNKS), 128, 0, stream>>>(x, p0, S, Q0);
  k_middle<<<1, 64, 0, stream>>>(S, Q0, M);
  k_pass2<<<dim3(BATCH, 49), 128, 0, stream>>>(x, M, out);
}